// Classification_4922032521468
// MI455X (gfx1250) — compile-verified
//
#include <hip/hip_runtime.h>

typedef float v2f __attribute__((ext_vector_type(2)));
typedef float v8f __attribute__((ext_vector_type(8)));

#define D_DIM 512   // per-token embedding dim
#define A_DIM 512   // number of activity tokens
#define C_DIM 4     // classes

// ---------------------------------------------------------------------------
// Phase 1: one f32 WMMA GEMM  [A x D] * [D x 8]  ->  LR[A][8]
//   cols 0..3 : e_a . W[c, 0:D]   + bias[c]   (left-operand contribution)
//   cols 4..7 : e_a . W[c, D:2D]              (right-operand contribution)
// One wave (32 threads) per 16-row M tile; V_WMMA_F32_16X16X4_F32 over K=D.
// B columns 8..15 are padding: they load valid-but-arbitrary W data and the
// resulting garbage stays confined to D columns 8..15 (WMMA columns are
// independent dot products), which the writeout discards. No masking needed.
// ---------------------------------------------------------------------------
__global__ __launch_bounds__(32) void lr_gemm_wmma_kernel(
    const float* __restrict__ embeds,          // [NUM_TOKENS, D]
    const int*   __restrict__ activity_index,  // [A]
    const float* __restrict__ W,               // [C, 2D]
    const float* __restrict__ bias,            // [C]
    float* __restrict__ LR)                    // [A, 8] workspace
{
    const int lane = threadIdx.x;   // wave32
    const int half = lane >> 4;     // 0: K sub 0..1, 1: K sub 2..3
    const int mn   = lane & 15;     // A-frag row / B-frag col
    const int kb   = half * 2;

    // A fragment: row mn of this 16-row tile of gathered activity embeddings.
    const int arow = blockIdx.x * 16 + mn;
    const int gidx = activity_index[arow];
    const float* aBase = embeds + (size_t)gidx * D_DIM;

    // B fragment: column n of the [D x 8] weight slab (cols 8..15 = don't-care).
    const int n  = mn;
    const int nc = n & 7; // clamped so every lane has a valid address
    const float* wBase = (nc < C_DIM)
        ? (W + (size_t)nc * (2 * D_DIM))                       // left half rows
        : (W + (size_t)(nc - C_DIM) * (2 * D_DIM) + D_DIM);    // right half rows

    v8f acc = {};
    for (int k = 0; k < D_DIM; k += 4) {
        v2f afrag = *(const v2f*)(aBase + k + kb);
        v2f bfrag = *(const v2f*)(wBase + k + kb);
        // 8 args: (neg_a, A, neg_b, B, c_mod, C, reuse_a, reuse_b)
        acc = __builtin_amdgcn_wmma_f32_16x16x4_f32(
            false, afrag, false, bfrag, (short)0, acc, false, false);
    }

    // D layout: VGPR r holds M = r (lanes 0-15) / M = 8+r (lanes 16-31), N = lane&15.
    if (n < 8) {
        const float badd = (n < C_DIM) ? bias[n] : 0.0f;  // fold bias into left cols
#pragma unroll
        for (int r = 0; r < 8; ++r) {
            const int m = half * 8 + r;
            LR[(size_t)(blockIdx.x * 16 + m) * 8 + n] = acc[r] + badd;
        }
    }
}

// ---------------------------------------------------------------------------
// Phase 2: for every pair (i < j): logits = LR[i][0:4] + LR[j][4:8];
// log_softmax over the 4 classes; store at the row-major triangular index.
// LR is 16 KB -> cache resident; this phase is pure streaming output.
// ---------------------------------------------------------------------------
__global__ __launch_bounds__(256) void pair_logsoftmax_kernel(
    const float* __restrict__ LR, float* __restrict__ out)
{
    const int j = blockIdx.x * blockDim.x + threadIdx.x;
    const int i = blockIdx.y;
    if (j >= A_DIM || j <= i) return;

    // pairs before row i: i*(2A - i - 1)/2  (matches jnp.triu_indices order)
    const size_t p = (size_t)i * (2 * A_DIM - i - 1) / 2 + (size_t)(j - i - 1);

    const float4 l = *(const float4*)(LR + (size_t)i * 8);      // left + bias
    const float4 r = *(const float4*)(LR + (size_t)j * 8 + 4);  // right

    const float x0 = l.x + r.x;
    const float x1 = l.y + r.y;
    const float x2 = l.z + r.z;
    const float x3 = l.w + r.w;

    const float m  = fmaxf(fmaxf(x0, x1), fmaxf(x2, x3));
    const float s  = __expf(x0 - m) + __expf(x1 - m) + __expf(x2 - m) + __expf(x3 - m);
    const float lse = __logf(s) + m;

    const float4 o = { x0 - lse, x1 - lse, x2 - lse, x3 - lse };
    *(float4*)(out + p * 4) = o;
}

extern "C" void kernel_launch(void* const* d_in, const int* in_sizes, int n_in,
                              void* d_out, int out_size, void* d_ws, size_t ws_size,
                              hipStream_t stream)
{
    const float* embeds = (const float*)d_in[0];   // [4096, 512] f32
    const int*   aidx   = (const int*)d_in[1];     // [512] int
    const float* W      = (const float*)d_in[2];   // [4, 1024] f32
    const float* bias   = (const float*)d_in[3];   // [4] f32
    float* LR  = (float*)d_ws;                     // [512, 8] f32 = 16 KB scratch
    float* out = (float*)d_out;                    // [130816, 4] f32

    // Phase 1: 32 M-tiles, one wave32 each -> v_wmma_f32_16x16x4_f32 path.
    lr_gemm_wmma_kernel<<<A_DIM / 16, 32, 0, stream>>>(embeds, aidx, W, bias, LR);

    // Phase 2: 512 x 512 (i, j) lattice; upper triangle produces output.
    dim3 g2((A_DIM + 255) / 256, A_DIM);
    pair_logsoftmax_kernel<<<g2, 256, 0, stream>>>(LR, out);
}